// DegreePrediction_11527692223142
// MI455X (gfx1250) — compile-verified
//
#include <hip/hip_runtime.h>
#include <cstdint>
#include <cstddef>

// Problem geometry (reference: N = 80, fixed).
#define NN 80
#define SLAB (NN * NN)            // 6400 floats per (s,t) slab, 25.6 KB
#define PAIRS_PER_BLOCK 4
#define NBLOCKS ((NN * NN) / PAIRS_PER_BLOCK)   // 1600 blocks
#define NTHREADS 320              // 10 wave32s; 16 (u) x 20 (v/4) layout

typedef __attribute__((ext_vector_type(4))) unsigned int u32x4;
typedef __attribute__((ext_vector_type(8))) int          i32x8;
typedef __attribute__((ext_vector_type(4))) int          i32x4;

// Issue one TDM load: 6400 contiguous f32 from global -> LDS.
// D# layout per CDNA5 ISA ch.8 (group0 128b, group1 256b, groups 2/3 zero).
// amdgpu-toolchain (clang-23) builtin is the 6-arg form:
//   (uint32x4 g0, int32x8 g1, int32x4 g2, int32x4 g3, int32x8 extra, i32 cpol)
__device__ __forceinline__ void tdm_load_slab(const float* gsrc,
                                              unsigned lds_byte_addr) {
  unsigned long long ga = (unsigned long long)(uintptr_t)gsrc;

  u32x4 g0;
  g0[0] = 1u;                                   // count=1 (valid), user mode
  g0[1] = lds_byte_addr;                        // LDS destination (bytes)
  g0[2] = (unsigned)(ga & 0xFFFFFFFFu);         // global_addr[31:0]
  g0[3] = (unsigned)((ga >> 32) & 0x01FFFFFFu)  // global_addr[56:32]
        | (2u << 30);                           // type = 2 ("image")

  i32x8 g1;
  g1[0] = (int)(2u << 16);                      // data_size = 2 -> 4 bytes
  g1[1] = (int)((unsigned)SLAB << 16);          // tensor_dim0[15:0] @ [63:48]
  g1[2] = (int)(((unsigned)SLAB >> 16)          // tensor_dim0[31:16]
        | (1u << 16));                          // tensor_dim1 = 1
  g1[3] = (int)((unsigned)SLAB << 16);          // tile_dim0 = 6400 @ [127:112]
  g1[4] = 1;                                    // tile_dim1 = 1, tile_dim2 = 0
  g1[5] = SLAB;                                 // tensor_dim0_stride = 6400
  g1[6] = 0;
  g1[7] = 0;

  i32x4 g2 = {0, 0, 0, 0};                      // <=2D: groups 2/3 unused
  i32x4 g3 = {0, 0, 0, 0};
  i32x8 g4 = {0, 0, 0, 0, 0, 0, 0, 0};          // extra group (zero-filled)

  __builtin_amdgcn_tensor_load_to_lds(g0, g1, g2, g3, g4, /*cpol=*/0);
}

__global__ __launch_bounds__(NTHREADS)
void degree_pred_kernel(const float* __restrict__ x,
                        const float* __restrict__ rz,
                        const float* __restrict__ rc,
                        const float* __restrict__ wt,
                        const float* __restrict__ wr,
                        float* __restrict__ y) {
  extern __shared__ float smem[];   // [2 buffers][3 arrays][SLAB]
  __shared__ float red[NN];         // per-block y partials

  const int tid   = threadIdx.x;
  const int pbase = blockIdx.x * PAIRS_PER_BLOCK;

  // Low 32 bits of an LDS generic address == LDS byte offset (ISA 10.2).
  const unsigned lds_base = (unsigned)(size_t)(void*)smem;

  // Prologue: stage pair 0 into buffer 0 (one wave issues; TDM ignores EXEC).
  if (tid == 0) {
    const size_t po = (size_t)pbase * SLAB;
    tdm_load_slab(wr + po, lds_base);
    tdm_load_slab(rz + po, lds_base + 1u * SLAB * 4u);
    tdm_load_slab(rc + po, lds_base + 2u * SLAB * 4u);
  }

  const int u0 = tid / 20;          // 0..15
  const int v4 = tid % 20;          // 0..19 -> v = 4*v4 .. 4*v4+3
  float acc[5] = {0.f, 0.f, 0.f, 0.f, 0.f};

  for (int i = 0; i < PAIRS_PER_BLOCK; ++i) {
    const int cur = i & 1;

    // Kick off DMA for the next pair into the other buffer (its last
    // consumers finished before the previous iteration's barrier).
    if ((i + 1) < PAIRS_PER_BLOCK && tid == 0) {
      const unsigned nb = lds_base + (unsigned)(1 - cur) * 3u * SLAB * 4u;
      const size_t po = (size_t)(pbase + i + 1) * SLAB;
      tdm_load_slab(wr + po, nb);
      tdm_load_slab(rz + po, nb + 1u * SLAB * 4u);
      tdm_load_slab(rc + po, nb + 2u * SLAB * 4u);
    }

    // TDM completes in order: allow the 3 newest to stay in flight.
    if ((i + 1) < PAIRS_PER_BLOCK) {
      __builtin_amdgcn_s_wait_tensorcnt(3);
    } else {
      __builtin_amdgcn_s_wait_tensorcnt(0);
    }
    __syncthreads();                // current buffer is now valid for all waves

    const int p = pbase + i;        // uniform -> scalar loads
    const float l2 = x[p] * wt[p];

    const float* swr = smem + cur * 3 * SLAB;
    const float* srz = swr + SLAB;
    const float* src = srz + SLAB;

#pragma unroll
    for (int k = 0; k < 5; ++k) {
      const int base = (u0 + 16 * k) * NN + v4 * 4;
      float4 a = *reinterpret_cast<const float4*>(swr + base);
      float4 b = *reinterpret_cast<const float4*>(srz + base);
      float4 c = *reinterpret_cast<const float4*>(src + base);
      float s = (fmaf(a.x, b.x, c.x) + fmaf(a.y, b.y, c.y))
              + (fmaf(a.z, b.z, c.z) + fmaf(a.w, b.w, c.w));
      acc[k] = fmaf(l2, s, acc[k]);
    }
    __syncthreads();                // done reading cur before it is re-DMA'd
  }

  // Block-level reduction over the 20 v-threads per u, then to global y.
  if (tid < NN) red[tid] = 0.f;
  __syncthreads();
#pragma unroll
  for (int k = 0; k < 5; ++k) {
    atomicAdd(&red[u0 + 16 * k], acc[k]);   // ds_add_f32
  }
  __syncthreads();
  if (tid < NN) atomicAdd(&y[tid], red[tid]);
}

__global__ void zero_out_kernel(float* __restrict__ y) {
  if (threadIdx.x < NN) y[threadIdx.x] = 0.f;
}

extern "C" void kernel_launch(void* const* d_in, const int* in_sizes, int n_in,
                              void* d_out, int out_size, void* d_ws,
                              size_t ws_size, hipStream_t stream) {
  // setup_inputs order: x, r_zeros, r_const, weights_t, weights_r (all f32)
  const float* x  = (const float*)d_in[0];
  const float* rz = (const float*)d_in[1];
  const float* rc = (const float*)d_in[2];
  const float* wt = (const float*)d_in[3];
  const float* wr = (const float*)d_in[4];
  float* y = (float*)d_out;

  (void)in_sizes; (void)n_in; (void)out_size; (void)d_ws; (void)ws_size;

  // d_out is poisoned by the harness; all blocks atomically accumulate,
  // so zero it first (stream-ordered, graph-capture safe).
  zero_out_kernel<<<1, 128, 0, stream>>>(y);

  const size_t shmem = (size_t)2 * 3 * SLAB * sizeof(float);  // 150 KB / WGP's 320 KB
  degree_pred_kernel<<<NBLOCKS, NTHREADS, shmem, stream>>>(x, rz, rc, wt, wr, y);
}